// EGConvLayer_76828374991621
// MI455X (gfx1250) — compile-verified
//
#include <hip/hip_runtime.h>

#define N_NODES 50000
#define N_EDGES 800000
#define K_IN    512
#define OUT_DIM 512
#define BFH     256            // B*FH = 4*64
#define NCOLS   800            // 256 (bases) + 32 (comb) + 512 (res)
#define EPSLN   1e-5f

typedef float v2f __attribute__((ext_vector_type(2)));
typedef float v8f __attribute__((ext_vector_type(8)));

// ---------------- weight concatenation: Wcat[k][j], j in [0,800) ----------------
__global__ void pack_w_kernel(const float* __restrict__ Wb, const float* __restrict__ Wc,
                              const float* __restrict__ Wr, float* __restrict__ wcat) {
    int idx = blockIdx.x * 256 + threadIdx.x;
    if (idx >= K_IN * NCOLS) return;
    int k = idx / NCOLS, j = idx % NCOLS;
    float v;
    if (j < BFH)            v = Wb[k * BFH + j];
    else if (j < BFH + 32)  v = Wc[k * 32 + (j - BFH)];
    else                    v = Wr[k * OUT_DIM + (j - BFH - 32)];
    wcat[idx] = v;
}

// ---------------- degree (float; also reused as integer in-degree source) ------
__global__ void deg_init_kernel(float* deg) {
    int i = blockIdx.x * 256 + threadIdx.x;
    if (i < N_NODES) deg[i] = 1.0f;                 // self-loop
}
__global__ void deg_accum_kernel(const int* __restrict__ ei, float* deg) {
    int e = blockIdx.x * 256 + threadIdx.x;
    if (e < N_EDGES) atomicAdd(&deg[ei[N_EDGES + e]], 1.0f);
}
__global__ void deg_inv_kernel(float* deg) {
    int i = blockIdx.x * 256 + threadIdx.x;
    if (i < N_NODES) deg[i] = rsqrtf(deg[i]);       // deg >= 1 always
}

// ---------------- CSR row offsets: single-block chunked scan -------------------
__global__ __launch_bounds__(256)
void csr_scan_kernel(const float* __restrict__ deg, int* __restrict__ offs) {
    __shared__ int lds[256];
    __shared__ int carry_s;
    int t = threadIdx.x;
    if (t == 0) carry_s = 0;
    __syncthreads();
    for (int base = 0; base < N_NODES; base += 256) {
        int i = base + t;
        int v = (i < N_NODES) ? ((int)deg[i] - 1) : 0;   // in-degree (deg holds indeg+1)
        lds[t] = v;
        __syncthreads();
#pragma unroll
        for (int off = 1; off < 256; off <<= 1) {
            int tv = (t >= off) ? lds[t - off] : 0;
            __syncthreads();
            lds[t] += tv;
            __syncthreads();
        }
        int incl  = lds[t];
        int carry = carry_s;
        if (i < N_NODES) offs[i] = carry + incl - v;     // exclusive prefix
        __syncthreads();
        if (t == 255) carry_s = carry + incl;            // chunk total
        __syncthreads();
    }
    if (t == 0) offs[N_NODES] = carry_s;                 // == N_EDGES
}

__global__ void cursor_init_kernel(const int* __restrict__ offs, int* __restrict__ cursor) {
    int i = blockIdx.x * 256 + threadIdx.x;
    if (i < N_NODES) cursor[i] = offs[i];
}

__global__ void csr_fill_kernel(const int* __restrict__ ei, int* __restrict__ cursor,
                                int* __restrict__ esrc) {
    int e = blockIdx.x * 256 + threadIdx.x;
    if (e >= N_EDGES) return;
    int src = ei[e];
    int dst = ei[N_EDGES + e];
    int pos = atomicAdd(&cursor[dst], 1);
    esrc[pos] = src;
}

// ---------------- fused GEMM: y[50000][800] = x @ [Wb|Wc|Wr] -------------------
// One wave -> 32x32 output tile: 2 A frags x 2 B frags = 4 WMMAs / k-step.
__global__ __launch_bounds__(256)
void gemm_wmma_kernel(const float* __restrict__ x, const float* __restrict__ wcat,
                      float* __restrict__ y) {
    const int MT  = N_NODES / 16;        // 3125 row tiles
    const int MT2 = (MT + 1) / 2;        // 1563 row-tile pairs
    const int NT  = NCOLS / 32;          // 25 col strips
    int wave = threadIdx.x >> 5;
    int lane = threadIdx.x & 31;
    int job  = blockIdx.x * 8 + wave;                 // wave-uniform
    if (job >= MT2 * NT) return;                      // uniform exit
    int mt2 = job / NT, nt = job % NT;
    int mt0  = mt2 * 2;
    int mt1  = (mt0 + 1 < MT) ? (mt0 + 1) : mt0;     // clamp: duplicate store, benign
    int col0 = nt * 32;

    int r  = lane & 15;                  // A row / B,C,D column
    int kh = lane >> 4;                  // k-half selector

    const float* xr0 = x + (size_t)(mt0 * 16 + r) * K_IN + 2 * kh;
    const float* xr1 = x + (size_t)(mt1 * 16 + r) * K_IN + 2 * kh;
    v8f acc00 = {}, acc01 = {}, acc10 = {}, acc11 = {};

#pragma unroll 4
    for (int k = 0; k < K_IN; k += 4) {
        v2f a0, a1, b0, b1;
        a0.x = xr0[k];  a0.y = xr0[k + 1];
        a1.x = xr1[k];  a1.y = xr1[k + 1];
        const float* brow = wcat + (size_t)(k + 2 * kh) * NCOLS + col0 + r;
        b0.x = brow[0];   b0.y = brow[NCOLS];
        b1.x = brow[16];  b1.y = brow[NCOLS + 16];
        acc00 = __builtin_amdgcn_wmma_f32_16x16x4_f32(false, a0, false, b0,
                                                      (short)0, acc00, false, false);
        acc01 = __builtin_amdgcn_wmma_f32_16x16x4_f32(false, a0, false, b1,
                                                      (short)0, acc01, false, false);
        acc10 = __builtin_amdgcn_wmma_f32_16x16x4_f32(false, a1, false, b0,
                                                      (short)0, acc10, false, false);
        acc11 = __builtin_amdgcn_wmma_f32_16x16x4_f32(false, a1, false, b1,
                                                      (short)0, acc11, false, false);
    }

    // C/D layout: VGPR i -> M = i + 8*kh, N = r
    float* yp0 = y + (size_t)(mt0 * 16) * NCOLS + col0 + r;
    float* yp1 = y + (size_t)(mt1 * 16) * NCOLS + col0 + r;
#pragma unroll
    for (int i = 0; i < 8; ++i) {
        int m = i + 8 * kh;
        yp0[(size_t)m * NCOLS]      = acc00[i];
        yp0[(size_t)m * NCOLS + 16] = acc01[i];
        yp1[(size_t)m * NCOLS]      = acc10[i];
        yp1[(size_t)m * NCOLS + 16] = acc11[i];
    }
}

// ---------------- CSR aggregation: one wave per node, no atomics ---------------
// acc = bases[node]*dinv^2 + sum_{src in-neighbors} bases[src]*dinv[src]*dinv[node]
__global__ __launch_bounds__(256)
void agg_csr_kernel(const float* __restrict__ y, const float* __restrict__ dinv,
                    const int* __restrict__ offs, const int* __restrict__ esrc,
                    float* __restrict__ agg) {
    int wave = threadIdx.x >> 5;
    int lane = threadIdx.x & 31;
    int node = blockIdx.x * 8 + wave;
    if (node >= N_NODES) return;

    float dn = dinv[node];
    const float* self = y + (size_t)node * NCOLS;    // bases in y[:, 0:256]
    float acc[8];
#pragma unroll
    for (int i = 0; i < 8; ++i) acc[i] = self[lane + 32 * i] * dn * dn;

    int p    = offs[node];
    int pend = offs[node + 1];

    // 2-deep software pipeline over incoming edges
    int srcp = 0;
    float cur[8];
#pragma unroll
    for (int i = 0; i < 8; ++i) cur[i] = 0.0f;
    if (p < pend) {
        srcp = esrc[p];
        const float* bs = y + (size_t)srcp * NCOLS;
#pragma unroll
        for (int i = 0; i < 8; ++i) cur[i] = bs[lane + 32 * i];
    }
    while (p < pend) {
        float nrm = dinv[srcp] * dn;
        int pn = p + 1;
        int srcn = 0;
        float nxt[8];
#pragma unroll
        for (int i = 0; i < 8; ++i) nxt[i] = 0.0f;
        if (pn < pend) {
            srcn = esrc[pn];
            const float* bs = y + (size_t)srcn * NCOLS;
#pragma unroll
            for (int i = 0; i < 8; ++i) nxt[i] = bs[lane + 32 * i];
        }
#pragma unroll
        for (int i = 0; i < 8; ++i) acc[i] = fmaf(cur[i], nrm, acc[i]);
#pragma unroll
        for (int i = 0; i < 8; ++i) cur[i] = nxt[i];
        srcp = srcn;
        p = pn;
    }

    float* ag = agg + (size_t)node * BFH;
#pragma unroll
    for (int i = 0; i < 8; ++i) ag[lane + 32 * i] = acc[i];
}

// ---------------- finalize: einsum + residual + LayerNorm + ReLU ---------------
__global__ __launch_bounds__(256)
void finalize_kernel(const float* __restrict__ y, const float* __restrict__ agg,
                     const float* __restrict__ b_comb, const float* __restrict__ conv_bias,
                     const float* __restrict__ b_res, const float* __restrict__ gamma,
                     const float* __restrict__ beta, float* __restrict__ out) {
    int node = blockIdx.x;
    int t = threadIdx.x;
    __shared__ float w[32];
    __shared__ float rs[8], rq[8];

    if (t < 32) w[t] = y[(size_t)node * NCOLS + BFH + t] + b_comb[t];
    __syncthreads();

    int f = t & 63;
    int h = t >> 6;                       // heads 0..3 (o0) / +4 (o1)
    const float* ag = agg + (size_t)node * BFH + f;
    float a0 = ag[0], a1 = ag[64], a2 = ag[128], a3 = ag[192];

    int o0 = t, o1 = t + 256;
    float v0 = w[h * 4 + 0] * a0 + w[h * 4 + 1] * a1 + w[h * 4 + 2] * a2 + w[h * 4 + 3] * a3;
    int h1 = h + 4;
    float v1 = w[h1 * 4 + 0] * a0 + w[h1 * 4 + 1] * a1 + w[h1 * 4 + 2] * a2 + w[h1 * 4 + 3] * a3;

    const float* res = y + (size_t)node * NCOLS + BFH + 32;
    v0 += conv_bias[o0] + res[o0] + b_res[o0];
    v1 += conv_bias[o1] + res[o1] + b_res[o1];

    float s = v0 + v1;
    float q = v0 * v0 + v1 * v1;
#pragma unroll
    for (int off = 16; off > 0; off >>= 1) {
        s += __shfl_down(s, off, 32);
        q += __shfl_down(q, off, 32);
    }
    int wv = t >> 5, ln = t & 31;
    if (ln == 0) { rs[wv] = s; rq[wv] = q; }
    __syncthreads();
    float S = 0.0f, Q = 0.0f;
#pragma unroll
    for (int i = 0; i < 8; ++i) { S += rs[i]; Q += rq[i]; }

    float mu   = S * (1.0f / 512.0f);
    float var  = Q * (1.0f / 512.0f) - mu * mu;
    float rstd = rsqrtf(var + EPSLN);

    float r0 = (v0 - mu) * rstd * gamma[o0] + beta[o0];
    float r1 = (v1 - mu) * rstd * gamma[o1] + beta[o1];
    out[(size_t)node * OUT_DIM + o0] = fmaxf(r0, 0.0f);
    out[(size_t)node * OUT_DIM + o1] = fmaxf(r1, 0.0f);
}

// ---------------- launch ----------------
extern "C" void kernel_launch(void* const* d_in, const int* in_sizes, int n_in,
                              void* d_out, int out_size, void* d_ws, size_t ws_size,
                              hipStream_t stream) {
    const float* x  = (const float*)d_in[0];
    const int*   ei = (const int*)  d_in[1];
    const float* Wb = (const float*)d_in[2];
    const float* Wc = (const float*)d_in[3];
    const float* bc = (const float*)d_in[4];
    const float* cb = (const float*)d_in[5];
    const float* Wr = (const float*)d_in[6];
    const float* br = (const float*)d_in[7];
    const float* gm = (const float*)d_in[8];
    const float* bt = (const float*)d_in[9];
    float* out = (float*)d_out;

    // scratch layout: Wcat | y | deg/dinv | agg | offs | cursor | esrc  (~217 MB)
    float* ws     = (float*)d_ws;
    float* wcat   = ws;
    float* y      = wcat + (size_t)K_IN * NCOLS;          // 409,600 f
    float* deg    = y + (size_t)N_NODES * NCOLS;          // +40,000,000 f
    float* agg    = deg + N_NODES;                        // +50,000 f
    int*   offs   = (int*)(agg + (size_t)N_NODES * BFH);  // +12,800,000 f
    int*   cursor = offs + (N_NODES + 1);
    int*   esrc   = cursor + N_NODES;

    pack_w_kernel   <<<(K_IN * NCOLS + 255) / 256, 256, 0, stream>>>(Wb, Wc, Wr, wcat);
    deg_init_kernel <<<(N_NODES + 255) / 256, 256, 0, stream>>>(deg);
    deg_accum_kernel<<<(N_EDGES + 255) / 256, 256, 0, stream>>>(ei, deg);
    csr_scan_kernel <<<1, 256, 0, stream>>>(deg, offs);
    cursor_init_kernel<<<(N_NODES + 255) / 256, 256, 0, stream>>>(offs, cursor);
    csr_fill_kernel <<<(N_EDGES + 255) / 256, 256, 0, stream>>>(ei, cursor, esrc);
    deg_inv_kernel  <<<(N_NODES + 255) / 256, 256, 0, stream>>>(deg);   // deg -> dinv

    const int jobs = ((N_NODES / 16 + 1) / 2) * (NCOLS / 32);           // 1563 * 25
    gemm_wmma_kernel<<<(jobs + 7) / 8, 256, 0, stream>>>(x, wcat, y);

    agg_csr_kernel  <<<(N_NODES + 7) / 8, 256, 0, stream>>>(y, deg, offs, esrc, agg);
    finalize_kernel <<<N_NODES, 256, 0, stream>>>(y, agg, bc, cb, br, gm, bt, out);
}